// LSHAttention_30872224923770
// MI455X (gfx1250) — compile-verified
//
#include <hip/hip_runtime.h>
#include <math.h>

// ---------------- problem constants (fixed by reference) ----------------
#define Bk   16
#define Ssz  4096
#define Dd   64
#define Hh   8
#define NBUC 512      // total buckets per batch = H * 64
#define Cch  512      // chunks per batch = H*S / 64
#define HS   32768    // H * S
#define LDK  66       // padded LDS stride for normalized keys (reg path)
#define LDV  68       // padded LDS stride for async-loaded rows (16B aligned)
#define LDP  130      // padded LDS stride for 128-col probs

typedef __attribute__((ext_vector_type(2))) float v2f;
typedef __attribute__((ext_vector_type(8))) float v8f;

// async copy of 16 bytes per lane: global (64-bit vaddr) -> LDS (32-bit offset)
__device__ __forceinline__ void async_g2l_b128(unsigned lds_off, const void* gaddr) {
    asm volatile("global_load_async_to_lds_b128 %0, %1, off"
                 :
                 : "v"(lds_off), "v"((unsigned long long)(uintptr_t)gaddr)
                 : "memory");
}
__device__ __forceinline__ void async_wait0() {
    asm volatile("s_wait_asynccnt 0x0" ::: "memory");
}

// ---------------- utility: zero fills ----------------
__global__ void zero_i32_kernel(int* p, int n) {
    int i = blockIdx.x * 256 + threadIdx.x;
    if (i < n) p[i] = 0;
}
__global__ void zero_f32_kernel(float* p, long long n) {
    long long i = (long long)blockIdx.x * 256 + threadIdx.x;
    if (i < n) p[i] = 0.f;
}

// ---------------- 1) LSH hashing: argmax over [r, -r] ----------------
// grid: (S/64, B), block: 64 threads (one token per thread)
__global__ __launch_bounds__(64) void hash_kernel(const float* __restrict__ qk,
                                                  const float* __restrict__ rot,
                                                  int* __restrict__ bucket_id) {
    __shared__ float rotl[64 * 256];   // rot[f][h*32+i], 64 KB
    __shared__ float qs[64 * 65];      // padded token rows
    int tid = threadIdx.x;
    int b = blockIdx.y;
    int t = blockIdx.x * 64 + tid;
    for (int i = tid; i < 64 * 256; i += 64) rotl[i] = rot[i];
    const float* qrow = qk + ((size_t)b * Ssz + t) * Dd;
    for (int d = 0; d < 64; d += 4) {
        float4 x = *(const float4*)(qrow + d);
        qs[tid * 65 + d + 0] = x.x; qs[tid * 65 + d + 1] = x.y;
        qs[tid * 65 + d + 2] = x.z; qs[tid * 65 + d + 3] = x.w;
    }
    __syncthreads();
    for (int h = 0; h < Hh; ++h) {
        float acc[32];
#pragma unroll
        for (int i = 0; i < 32; ++i) acc[i] = 0.f;
        for (int f = 0; f < 64; ++f) {
            float q = qs[tid * 65 + f];
            const float* rp = &rotl[f * 256 + h * 32];
#pragma unroll
            for (int i = 0; i < 32; ++i) acc[i] = fmaf(q, rp[i], acc[i]);
        }
        // argmax over concat([r, -r]); strict > keeps first occurrence
        float best = acc[0]; int bi = 0;
#pragma unroll
        for (int i = 1; i < 32; ++i) if (acc[i] > best) { best = acc[i]; bi = i; }
#pragma unroll
        for (int i = 0; i < 32; ++i) { float nv = -acc[i]; if (nv > best) { best = nv; bi = 32 + i; } }
        bucket_id[(size_t)b * HS + h * Ssz + t] = bi + (h << 6);
    }
}

// ---------------- 2) counting sort: histogram / prefix / stable scatter ----------------
__global__ void hist_kernel(const int* __restrict__ bucket_id, int* __restrict__ counts) {
    int i = blockIdx.x * 256 + threadIdx.x;  // over B*HS
    int b = i >> 15;
    atomicAdd(&counts[b * NBUC + bucket_id[i]], 1);
}

__global__ void prefix_kernel(const int* __restrict__ counts, int* __restrict__ offs) {
    if (threadIdx.x != 0) return;
    int b = blockIdx.x;
    int run = 0;
    for (int i = 0; i < NBUC; ++i) { offs[b * NBUC + i] = run; run += counts[b * NBUC + i]; }
}

// one thread per (b, bucket): stable scan over t ascending -> matches argsort of S*bucket + t
__global__ void scatter_kernel(const int* __restrict__ bucket_id, const int* __restrict__ offs,
                               int* __restrict__ st, int* __restrict__ undo) {
    int gid = blockIdx.x * 128 + threadIdx.x;  // over B*NBUC
    int b = gid >> 9;
    int bkt = gid & 511;
    int h = bkt >> 6;
    int base = offs[b * NBUC + bkt];
    const int* row = bucket_id + (size_t)b * HS + h * Ssz;
    for (int t = 0; t < Ssz; ++t) {
        if ((t & 63) == 0 && t + 512 < Ssz) __builtin_prefetch(row + t + 512, 0, 0);
        if (row[t] == bkt) {
            st[(size_t)b * HS + base] = t;
            undo[(size_t)b * HS + h * Ssz + t] = base;
            ++base;
        }
    }
}

// ---------------- 3/5) chunked attention via f32 WMMA ----------------
// grid: B*C blocks, 128 threads (4 waves). Wave w computes query rows [16w,16w+16).
// PASS 0: store per-row LSE. PASS 1: recompute probs, scale by hash weight, bo = P@V,
//         atomicAdd into out.
template <int PASS>
__global__ __launch_bounds__(128) void attn_kernel(const float* __restrict__ qk,
                                                   const float* __restrict__ vv,
                                                   const int* __restrict__ st,
                                                   float* __restrict__ slog,
                                                   const float* __restrict__ pw,
                                                   float* __restrict__ out) {
    __shared__ float bq[64 * LDV];    // raw queries       (async-loaded)
    __shared__ float bkl[128 * LDK];  // normalized keys   (register path)
    __shared__ float bvl[128 * LDV];  // values            (async-loaded)
    __shared__ float pr[64 * LDP];    // probs staging (PASS 1)
    __shared__ int   tkv[128];        // original positions
    __shared__ float wrow[64];        // per-row hash weight (PASS 1)

    int tid = threadIdx.x;
    int blk = blockIdx.x;
    int b = blk >> 9;
    int c = blk & 511;
    int h = c >> 6;
    int cprev = (c + Cch - 1) & (Cch - 1);   // look-one-back wraps mod C

    // ---- cooperative load: thread j owns kv row j ----
    int j = tid;
    int sidx = (j < 64) ? (c * 64 + j) : (cprev * 64 + (j - 64));
    int t = st[(size_t)b * HS + sidx];
    tkv[j] = t;
    const float* qrow = qk + ((size_t)b * Ssz + t) * Dd;

    // async DMA: values row j -> bvl (PASS 1 only)
    if (PASS == 1) {
        const float* vrow = vv + ((size_t)b * Ssz + t) * Dd;
        unsigned lds_bv = (unsigned)(uintptr_t)(&bvl[j * LDV]);
#pragma unroll
        for (int d = 0; d < 64; d += 4) async_g2l_b128(lds_bv + d * 4, vrow + d);
    }
    // async DMA: raw query rows -> bq, issued by upper half of the block
    if (j >= 64) {
        int jq = j - 64;
        int tq2 = st[(size_t)b * HS + c * 64 + jq];
        const float* q2 = qk + ((size_t)b * Ssz + tq2) * Dd;
        unsigned lds_bq = (unsigned)(uintptr_t)(&bq[jq * LDV]);
#pragma unroll
        for (int d = 0; d < 64; d += 4) async_g2l_b128(lds_bq + d * 4, q2 + d);
    }

    // register path: keys need a row norm, so stage through VGPRs
    float ssq = 0.f;
    for (int d = 0; d < 64; d += 4) {
        float4 x = *(const float4*)(qrow + d);
        ssq += x.x * x.x + x.y * x.y + x.z * x.z + x.w * x.w;
        bkl[j * LDK + d + 0] = x.x; bkl[j * LDK + d + 1] = x.y;
        bkl[j * LDK + d + 2] = x.z; bkl[j * LDK + d + 3] = x.w;
    }
    float scn = 1.f / fmaxf(sqrtf(ssq), 1e-12f);
    for (int d = 0; d < 64; ++d) bkl[j * LDK + d] *= scn;
    if (PASS == 1 && j < 64) wrow[j] = pw[(size_t)b * HS + h * Ssz + t];

    async_wait0();      // all my async LDS writes landed
    __syncthreads();    // visible block-wide

    int wv = tid >> 5;
    int lane = tid & 31;
    int half = lane >> 4;   // K half select for A/B operands
    int l16 = lane & 15;
    int arow = wv * 16 + l16;

    // ---- dots = bq (16x64) x bk^T (64x128), 8 N-tiles, K-steps of 4 ----
    v8f acc[8];
#pragma unroll
    for (int nt = 0; nt < 8; ++nt) {
        v8f ca = {0.f, 0.f, 0.f, 0.f, 0.f, 0.f, 0.f, 0.f};
#pragma unroll
        for (int k0 = 0; k0 < 64; k0 += 4) {
            int kk = k0 + half * 2;
            v2f a, bb;
            a[0] = bq[arow * LDV + kk];
            a[1] = bq[arow * LDV + kk + 1];
            bb[0] = bkl[(nt * 16 + l16) * LDK + kk];
            bb[1] = bkl[(nt * 16 + l16) * LDK + kk + 1];
            ca = __builtin_amdgcn_wmma_f32_16x16x4_f32(false, a, false, bb, (short)0, ca,
                                                       false, false);
        }
        acc[nt] = ca;
    }

    // ---- scale + self-mask. C layout: VGPR r, lane -> row r+8*half, col l16 ----
    int tq[8];
#pragma unroll
    for (int r = 0; r < 8; ++r) tq[r] = tkv[wv * 16 + r + 8 * half];
#pragma unroll
    for (int nt = 0; nt < 8; ++nt) {
        int tj = tkv[nt * 16 + l16];
#pragma unroll
        for (int r = 0; r < 8; ++r) {
            float dv = acc[nt][r] * 0.125f;           // D^-0.5
            acc[nt][r] = (tq[r] == tj) ? -5.0e4f : dv;
        }
    }

    // ---- row logsumexp: reduce over 8 tiles in-lane + 16-lane half via shfl_xor ----
    float rmax[8], rsum[8], lse[8];
#pragma unroll
    for (int r = 0; r < 8; ++r) {
        float m = -3.0e38f;
#pragma unroll
        for (int nt = 0; nt < 8; ++nt) m = fmaxf(m, acc[nt][r]);
        rmax[r] = m;
    }
    for (int off = 1; off < 16; off <<= 1) {
#pragma unroll
        for (int r = 0; r < 8; ++r) rmax[r] = fmaxf(rmax[r], __shfl_xor(rmax[r], off, 32));
    }
#pragma unroll
    for (int r = 0; r < 8; ++r) {
        float s = 0.f;
#pragma unroll
        for (int nt = 0; nt < 8; ++nt) s += expf(acc[nt][r] - rmax[r]);
        rsum[r] = s;
    }
    for (int off = 1; off < 16; off <<= 1) {
#pragma unroll
        for (int r = 0; r < 8; ++r) rsum[r] += __shfl_xor(rsum[r], off, 32);
    }
#pragma unroll
    for (int r = 0; r < 8; ++r) lse[r] = logf(rsum[r]) + rmax[r];

    if (PASS == 0) {
        if (l16 == 0) {
#pragma unroll
            for (int r = 0; r < 8; ++r)
                slog[(size_t)b * HS + c * 64 + wv * 16 + r + 8 * half] = lse[r];
        }
        return;
    }

    // ---- PASS 1: probs -> LDS, bo = probs (16x128) x bv (128x64) ----
#pragma unroll
    for (int nt = 0; nt < 8; ++nt) {
#pragma unroll
        for (int r = 0; r < 8; ++r)
            pr[(wv * 16 + r + 8 * half) * LDP + nt * 16 + l16] = expf(acc[nt][r] - lse[r]);
    }
    __syncthreads();

#pragma unroll
    for (int dt = 0; dt < 4; ++dt) {
        v8f oc = {0.f, 0.f, 0.f, 0.f, 0.f, 0.f, 0.f, 0.f};
#pragma unroll
        for (int k0 = 0; k0 < 128; k0 += 4) {
            int kk = k0 + half * 2;
            v2f a, bb;
            a[0] = pr[arow * LDP + kk];
            a[1] = pr[arow * LDP + kk + 1];
            bb[0] = bvl[kk * LDV + dt * 16 + l16];
            bb[1] = bvl[(kk + 1) * LDV + dt * 16 + l16];
            oc = __builtin_amdgcn_wmma_f32_16x16x4_f32(false, a, false, bb, (short)0, oc,
                                                       false, false);
        }
#pragma unroll
        for (int r = 0; r < 8; ++r) {
            int m = wv * 16 + r + 8 * half;
            float w = wrow[m];
            atomicAdd(&out[((size_t)b * Ssz + tq[r]) * Dd + dt * 16 + l16], w * oc[r]);
        }
    }
}

// ---------------- 4) per-(b,t) hash-combine weights ----------------
__global__ void weights_kernel(const float* __restrict__ slog, const int* __restrict__ undo,
                               float* __restrict__ pw) {
    int i = blockIdx.x * 256 + threadIdx.x;  // over B*S
    int b = i >> 12;
    int t = i & 4095;
    float l[8];
    float m = -3.0e38f;
#pragma unroll
    for (int h = 0; h < Hh; ++h) {
        l[h] = slog[(size_t)b * HS + undo[(size_t)b * HS + h * Ssz + t]];
        m = fmaxf(m, l[h]);
    }
    float ssum = 0.f;
#pragma unroll
    for (int h = 0; h < Hh; ++h) ssum += expf(l[h] - m);
    float lt = logf(ssum) + m;
#pragma unroll
    for (int h = 0; h < Hh; ++h) pw[(size_t)b * HS + h * Ssz + t] = expf(l[h] - lt);
}

// ---------------- host launch ----------------
extern "C" void kernel_launch(void* const* d_in, const int* in_sizes, int n_in,
                              void* d_out, int out_size, void* d_ws, size_t ws_size,
                              hipStream_t stream) {
    (void)in_sizes; (void)n_in; (void)out_size; (void)ws_size;
    const float* qk  = (const float*)d_in[0];
    const float* v   = (const float*)d_in[1];
    const float* rot = (const float*)d_in[2];
    float* out = (float*)d_out;

    char* w = (char*)d_ws;
    int*   bucket_id = (int*)w;   w += (size_t)Bk * HS * 4;
    int*   counts    = (int*)w;   w += (size_t)Bk * NBUC * 4;
    int*   offs      = (int*)w;   w += (size_t)Bk * NBUC * 4;
    int*   st        = (int*)w;   w += (size_t)Bk * HS * 4;
    int*   undo      = (int*)w;   w += (size_t)Bk * HS * 4;
    float* slog      = (float*)w; w += (size_t)Bk * HS * 4;
    float* pw        = (float*)w; w += (size_t)Bk * HS * 4;

    zero_i32_kernel<<<(Bk * NBUC + 255) / 256, 256, 0, stream>>>(counts, Bk * NBUC);
    hash_kernel<<<dim3(Ssz / 64, Bk), 64, 0, stream>>>(qk, rot, bucket_id);
    hist_kernel<<<(Bk * HS) / 256, 256, 0, stream>>>(bucket_id, counts);
    prefix_kernel<<<Bk, 32, 0, stream>>>(counts, offs);
    scatter_kernel<<<(Bk * NBUC) / 128, 128, 0, stream>>>(bucket_id, offs, st, undo);
    attn_kernel<0><<<Bk * Cch, 128, 0, stream>>>(qk, v, st, slog, pw, out);
    weights_kernel<<<(Bk * Ssz) / 256, 256, 0, stream>>>(slog, undo, pw);
    long long outn = (long long)Bk * Ssz * Dd;
    zero_f32_kernel<<<(int)((outn + 255) / 256), 256, 0, stream>>>(out, outn);
    attn_kernel<1><<<Bk * Cch, 128, 0, stream>>>(qk, v, st, slog, pw, out);
}